// D4_Pool_28226525069638
// MI455X (gfx1250) — compile-verified
//
#include <hip/hip_runtime.h>
#include <cstdint>
#include <cstddef>

// D4-pool: out[g,c,h,w] = mean over 8 un-done orientations of x[(g*8+k),c,:,:]
//   k0: in[h][w]         k2: in[63-h][63-w]   k4: in[h][63-w]   k6: in[63-h][w]
//   k1: in[w][63-h]      k3: in[63-w][h]      k5: in[w][h]      k7: in[63-w][63-h]
// Direct orientations (0,2,4,6) are row-coalesced -> float4 global loads.
// Transposed orientations (1,3,5,7) are staged via async global->LDS DMA
// (ASYNCcnt path, CDNA5) into a bank-conflict-free padded LDS tile.

typedef float f4 __attribute__((ext_vector_type(4)));

typedef __attribute__((address_space(1))) int g_int;
typedef __attribute__((address_space(3))) int l_int;

#define TPB 256
#define NN  64
#define NP  65   // padded row: stride 65 -> transposed column reads are conflict-free

#if defined(__gfx1250__) && __has_builtin(__builtin_amdgcn_global_load_async_to_lds_b32)
#define USE_ASYNC 1
#else
#define USE_ASYNC 0
#endif

__device__ __forceinline__ void stage_plane(const float* __restrict__ src,
                                            float (*ldsp)[NP], int tid) {
#if USE_ASYNC
  // 64x64 plane, lane-contiguous b32 async copies: each wave op moves 128B.
  #pragma unroll
  for (int it = 0; it < 16; ++it) {
    int q = it * TPB + tid;            // flat float index 0..4095
    int r = q >> 6, c = q & 63;
    __builtin_amdgcn_global_load_async_to_lds_b32(
        (g_int*)(src + q),
        (l_int*)&ldsp[r][c],
        /*offset=*/0, /*cpol=*/0);
  }
#else
  // Fallback: coalesced float4 global loads + scalar LDS stores.
  #pragma unroll
  for (int it = 0; it < 4; ++it) {
    int q = it * TPB + tid;            // float4 chunk 0..1023
    int r = q >> 4, c = (q & 15) << 2;
    f4 v = *(const f4*)(src + (size_t)q * 4);
    ldsp[r][c + 0] = v[0];
    ldsp[r][c + 1] = v[1];
    ldsp[r][c + 2] = v[2];
    ldsp[r][c + 3] = v[3];
  }
#endif
}

__device__ __forceinline__ void wait_async_done() {
#if USE_ASYNC
#if __has_builtin(__builtin_amdgcn_s_wait_asynccnt)
  __builtin_amdgcn_s_wait_asynccnt(0);
#else
  asm volatile("s_wait_asynccnt 0" ::: "memory");
#endif
#endif
}

__global__ __launch_bounds__(TPB) void d4_pool_kernel(const float* __restrict__ x,
                                                      float* __restrict__ out) {
  // Shapes fixed by the reference: B=256, C=128, H=W=64, G=32.
  const int gc  = blockIdx.x;          // g*128 + c, 0..4095 (one output plane)
  const int g   = gc >> 7;
  const int c   = gc & 127;
  const int tid = threadIdx.x;

  const size_t plane   = 64 * 64;             // 4096 floats
  const size_t kstride = (size_t)128 * plane; // stride between orientations k
  const float* base = x + ((size_t)g * 8 * 128 + c) * plane;  // k=0 plane

  __shared__ float lds[2][NN][NP];           // 2 staged planes, padded rows

  f4 acc[4];

  // ----------------- Phase 0: k=0,2 direct; k=1,3 via LDS -----------------
  stage_plane(base + 1 * kstride, lds[0], tid);
  stage_plane(base + 3 * kstride, lds[1], tid);

  #pragma unroll
  for (int it = 0; it < 4; ++it) {
    int q   = it * TPB + tid;
    int oh  = q >> 4;
    int ow0 = (q & 15) << 2;
    f4 v0 = __builtin_nontemporal_load((const f4*)(base + (size_t)oh * 64 + ow0));
    f4 v2 = __builtin_nontemporal_load(
        (const f4*)(base + 2 * kstride + (size_t)(63 - oh) * 64 + (60 - ow0)));
    f4 s = v0;
    #pragma unroll
    for (int j = 0; j < 4; ++j) s[j] += v2[3 - j];   // reverse within row
    acc[it] = s;
  }

  wait_async_done();
  __syncthreads();

  #pragma unroll
  for (int it = 0; it < 4; ++it) {
    int q   = it * TPB + tid;
    int oh  = q >> 4;
    int ow0 = (q & 15) << 2;
    #pragma unroll
    for (int j = 0; j < 4; ++j) {
      int ow = ow0 + j;
      acc[it][j] += lds[0][ow][63 - oh]        // k1: in[w][63-h]
                  + lds[1][63 - ow][oh];       // k3: in[63-w][h]
    }
  }

  __syncthreads();   // all reads done before LDS is overwritten

  // ----------------- Phase 1: k=4,6 direct; k=5,7 via LDS -----------------
  stage_plane(base + 5 * kstride, lds[0], tid);
  stage_plane(base + 7 * kstride, lds[1], tid);

  #pragma unroll
  for (int it = 0; it < 4; ++it) {
    int q   = it * TPB + tid;
    int oh  = q >> 4;
    int ow0 = (q & 15) << 2;
    f4 v4 = __builtin_nontemporal_load(
        (const f4*)(base + 4 * kstride + (size_t)oh * 64 + (60 - ow0)));
    f4 v6 = __builtin_nontemporal_load(
        (const f4*)(base + 6 * kstride + (size_t)(63 - oh) * 64 + ow0));
    f4 s = acc[it] + v6;
    #pragma unroll
    for (int j = 0; j < 4; ++j) s[j] += v4[3 - j];
    acc[it] = s;
  }

  wait_async_done();
  __syncthreads();

  float* oplane = out + (size_t)gc * plane;
  #pragma unroll
  for (int it = 0; it < 4; ++it) {
    int q   = it * TPB + tid;
    int oh  = q >> 4;
    int ow0 = (q & 15) << 2;
    f4 s = acc[it];
    #pragma unroll
    for (int j = 0; j < 4; ++j) {
      int ow = ow0 + j;
      s[j] += lds[0][ow][oh]                   // k5: in[w][h]
            + lds[1][63 - ow][63 - oh];        // k7: in[63-w][63-h]
    }
    s *= 0.125f;
    __builtin_nontemporal_store(s, (f4*)(oplane + (size_t)oh * 64 + ow0));
  }
}

extern "C" void kernel_launch(void* const* d_in, const int* in_sizes, int n_in,
                              void* d_out, int out_size, void* d_ws, size_t ws_size,
                              hipStream_t stream) {
  (void)in_sizes; (void)n_in; (void)d_ws; (void)ws_size;
  const float* x = (const float*)d_in[0];
  float* out     = (float*)d_out;
  // out_size = G*C*H*W = 32*128*4096 -> 4096 output planes, one block each.
  int nplanes = out_size >> 12;
  d4_pool_kernel<<<dim3(nplanes), dim3(TPB), 0, stream>>>(x, out);
}